// Encoder_12257836662966
// MI455X (gfx1250) — compile-verified
//
#include <hip/hip_runtime.h>
#include <hip/hip_bf16.h>

#define NNODES 50000
#define NEDGES 1600000
#define DIM    128

typedef __attribute__((ext_vector_type(2))) float v2f;
typedef __attribute__((ext_vector_type(8))) float v8f;

// ---------------------------------------------------------------- utilities
__global__ void zero_kernel(float* __restrict__ p, int n) {
    int i = blockIdx.x * blockDim.x + threadIdx.x;
    if (i < n) p[i] = 0.0f;
}

// deg[dst] += 1 for every edge (self-loop's +1 added in dinv_kernel)
__global__ void deg_kernel(const int* __restrict__ dst, float* __restrict__ deg, int e) {
    int i = blockIdx.x * blockDim.x + threadIdx.x;
    if (i < e) atomicAdd(&deg[dst[i]], 1.0f);
}

// in-place: deg -> dinv = rsqrt(deg + 1)   (+1 = self loop; always > 0)
__global__ void dinv_kernel(float* __restrict__ d, int n) {
    int i = blockIdx.x * blockDim.x + threadIdx.x;
    if (i < n) d[i] = rsqrtf(d[i] + 1.0f);
}

// ---------------------------------------------------------------- GEMM (WMMA)
// out[NNODES x DIM] = A[NNODES x DIM] @ W[DIM x DIM], fp32 WMMA 16x16x4.
// One wave -> one 16x16 output tile. 8 waves/block cover the 128 columns.
// blockIdx.x selects the 16-row band. NNODES % 16 == 0 -> no partial tiles,
// EXEC stays all-ones as required for V_WMMA.
__global__ __launch_bounds__(256) void gemm_wmma_kernel(const float* __restrict__ A,
                                                        const float* __restrict__ W,
                                                        float* __restrict__ out) {
    const int lane = threadIdx.x & 31;
    const int wave = threadIdx.x >> 5;
    const int m0   = blockIdx.x * 16;
    const int n0   = wave * 16;
    const int hi   = lane >> 4;     // half-wave select
    const int l15  = lane & 15;

    // A-matrix 16x4 fp32 layout: lane l15 holds row M=l15;
    //   lanes 0-15:  VGPR0=K0, VGPR1=K1 ; lanes 16-31: VGPR0=K2, VGPR1=K3
    const float* arow = A + (size_t)(m0 + l15) * DIM + 2 * hi;
    // B-matrix 4x16 fp32: VGPR0 = row (k+2*hi), VGPR1 = row (k+1+2*hi), N=n0+l15
    const float* wcol = W + (size_t)(2 * hi) * DIM + n0 + l15;

    v8f c = {};
#pragma unroll 8
    for (int k = 0; k < DIM; k += 4) {
        float2 av = *reinterpret_cast<const float2*>(arow + k);   // K=k+2hi, k+1+2hi
        v2f a; a.x = av.x; a.y = av.y;
        v2f b;
        b.x = wcol[(size_t)k * DIM];
        b.y = wcol[(size_t)(k + 1) * DIM];
        // 8 args: (neg_a, A, neg_b, B, c_mod, C, reuse_a, reuse_b)
        c = __builtin_amdgcn_wmma_f32_16x16x4_f32(false, a, false, b,
                                                  (short)0, c, false, false);
    }

    // C/D layout: VGPR r -> M = r + 8*hi, N = l15
    float* orow = out + (size_t)(m0 + 8 * hi) * DIM + n0 + l15;
#pragma unroll
    for (int r = 0; r < 8; ++r) orow[(size_t)r * DIM] = c[r];
}

// ---------------------------------------------------------------- edge scatter
// One wave per edge, one float4 per lane (128 floats = 32 lanes * 4).
// agg[dst] += h[src] * dinv[src] * dinv[dst]
__global__ __launch_bounds__(256) void scatter_kernel(const float* __restrict__ h,
                                                      const int* __restrict__ src,
                                                      const int* __restrict__ dst,
                                                      const float* __restrict__ dinv,
                                                      float* __restrict__ agg, int e) {
    int tid  = blockIdx.x * blockDim.x + threadIdx.x;
    int edge = tid >> 5;
    int lane = tid & 31;
    if (edge >= e) return;
    int s = src[edge];
    int d = dst[edge];
    float w = dinv[s] * dinv[d];
    float4 v = reinterpret_cast<const float4*>(h + (size_t)s * DIM)[lane];
    float* ap = agg + (size_t)d * DIM + lane * 4;
    atomicAdd(ap + 0, v.x * w);
    atomicAdd(ap + 1, v.y * w);
    atomicAdd(ap + 2, v.z * w);
    atomicAdd(ap + 3, v.w * w);
}

// ---------------------------------------------------------------- epilogue
// out = relu(agg + h*dinv^2 (self-loop) + bias). Safe in-place (agg==out).
__global__ void epi_kernel(const float* __restrict__ agg, const float* __restrict__ h,
                           const float* __restrict__ dinv, const float* __restrict__ b,
                           float* __restrict__ out, int nd) {
    int t = blockIdx.x * blockDim.x + threadIdx.x;
    if (t >= nd) return;
    int i = t >> 7;        // node
    int j = t & 127;       // feature
    float di = dinv[i];
    float v = agg[t] + h[t] * di * di + b[j];
    out[t] = fmaxf(v, 0.0f);
}

// ---------------------------------------------------------------- launcher
extern "C" void kernel_launch(void* const* d_in, const int* in_sizes, int n_in,
                              void* d_out, int out_size, void* d_ws, size_t ws_size,
                              hipStream_t stream) {
    (void)in_sizes; (void)n_in; (void)out_size; (void)ws_size;

    const float* x  = (const float*)d_in[0];
    const int*   ei = (const int*)d_in[1];   // (2, E) row-major
    const float* W0 = (const float*)d_in[2];
    const float* b0 = (const float*)d_in[3];
    const float* W1 = (const float*)d_in[4];
    const float* b1 = (const float*)d_in[5];
    float* out = (float*)d_out;

    const int* src = ei;
    const int* dst = ei + NEDGES;

    const int ND = NNODES * DIM;
    float* ws   = (float*)d_ws;
    float* dinv = ws;                 // N floats (deg, then dinv in-place)
    float* h    = ws + 50176;         // N*D floats (aligned past dinv)
    float* agg  = h + ND;             // N*D floats

    const int B = 256;
    dim3 blk(B);
    dim3 gN((NNODES + B - 1) / B);
    dim3 gE((NEDGES + B - 1) / B);
    dim3 gND((ND + B - 1) / B);
    dim3 gScat((NEDGES * 32 + B - 1) / B);   // wave per edge
    dim3 gGemm(NNODES / 16);                  // exact: 3125 row tiles

    // normalization: deg -> dinv
    zero_kernel<<<gN, blk, 0, stream>>>(dinv, NNODES);
    deg_kernel<<<gE, blk, 0, stream>>>(dst, dinv, NEDGES);
    dinv_kernel<<<gN, blk, 0, stream>>>(dinv, NNODES);

    // ---- layer 1 ----
    gemm_wmma_kernel<<<gGemm, blk, 0, stream>>>(x, W0, h);
    zero_kernel<<<gND, blk, 0, stream>>>(agg, ND);
    scatter_kernel<<<gScat, blk, 0, stream>>>(h, src, dst, dinv, agg, NEDGES);
    epi_kernel<<<gND, blk, 0, stream>>>(agg, h, dinv, b0, agg, ND);  // agg <- relu(...)

    // ---- layer 2 ----
    gemm_wmma_kernel<<<gGemm, blk, 0, stream>>>(agg, W1, h);
    zero_kernel<<<gND, blk, 0, stream>>>(agg, ND);
    scatter_kernel<<<gScat, blk, 0, stream>>>(h, src, dst, dinv, agg, NEDGES);
    epi_kernel<<<gND, blk, 0, stream>>>(agg, h, dinv, b1, out, ND);
}